// PositionalEncoding_84095459656362
// MI455X (gfx1250) — compile-verified
//
#include <hip/hip_runtime.h>
#include <hip/hip_bf16.h>

// PositionalEncoding: out[b,s,d] = x[b,s,d] + pe[s,d]
//   pe[s,i] = sin(s / 10000^((i/2)/S)) if i even else cos(...)
// B=8, S=4096, D=1024, fp32.  Memory-bound: 268 MB @ 23.3 TB/s ~= 11.5 us.
// Strategy: issue first async global->LDS b128 loads immediately (overlap HBM
// latency with PE transcendentals), compute pe once per (s,d) thread and reuse
// across the 8 batches, stream x through a triple-buffered LDS pipeline
// (per-thread-private LDS regions => only s_wait_asynccnt, no barriers),
// nontemporal b128 stores.

#define PB 8
#define PS 4096
#define PD 1024

typedef __attribute__((ext_vector_type(4))) float f32x4;

// Builtin expects: (int __vector_size(16) addrspace(1)*, ... addrspace(3)*, imm, imm)
typedef int b128_t __attribute__((vector_size(16)));
typedef __attribute__((address_space(1))) b128_t* gp128;
typedef __attribute__((address_space(3))) b128_t* lp128;

#if defined(__has_builtin)
#if __has_builtin(__builtin_amdgcn_global_load_async_to_lds_b128)
#define HAVE_ASYNC_B128 1
#endif
#if __has_builtin(__builtin_amdgcn_s_wait_asynccnt)
#define HAVE_WAIT_ASYNC 1
#endif
#endif

__device__ __forceinline__ void async_load16(const float* gsrc, float* lds_dst) {
#if defined(HAVE_ASYNC_B128)
  __builtin_amdgcn_global_load_async_to_lds_b128(
      (gp128)gsrc, (lp128)lds_dst, /*offset=*/0, /*cpol=*/0);
#else
  unsigned lds_off = (unsigned)(size_t)lds_dst;
  asm volatile("global_load_async_to_lds_b128 %0, %1, off"
               :: "v"(lds_off), "v"(gsrc)
               : "memory");
#endif
}

template <int N>
__device__ __forceinline__ void wait_async() {
#if defined(HAVE_WAIT_ASYNC)
  __builtin_amdgcn_s_wait_asynccnt(N);
#else
  asm volatile("s_wait_asynccnt %0" :: "i"(N) : "memory");
#endif
}

// n is a compile-time constant after full unroll; folds to one s_wait_asynccnt.
__device__ __forceinline__ void wait_async_n(int n) {
  if (n >= 2)      wait_async<2>();
  else if (n == 1) wait_async<1>();
  else             wait_async<0>();
}

__global__ __launch_bounds__(256) void pe_add_kernel(const float* __restrict__ x,
                                                     float* __restrict__ out) {
  __shared__ __align__(16) float buf[3][PD];  // 12 KB triple buffer

  const int s  = blockIdx.x;    // sequence position
  const int t  = threadIdx.x;   // 0..255, owns d = 4t..4t+3
  const int d0 = t * 4;

  const size_t row     = (size_t)s * PD + (size_t)d0;  // offset inside one batch
  const size_t bstride = (size_t)PS * PD;              // 4M elements = 16 MiB

  // ---- kick off the pipeline first: HBM latency overlaps PE transcendentals
  async_load16(x + row,           &buf[0][d0]);
  async_load16(x + row + bstride, &buf[1][d0]);

  // ---- positional encoding for this thread's 4 lanes (reused over batch) ----
  // inv_freq(k) = 10000^(-k/S) = exp2(-k * log2(10000)/S)
  const float kscale = 13.287712379549449f / (float)PS;  // log2(10000)/S
  const float a0 = (float)s * exp2f(-(float)(2 * t) * kscale);
  const float a1 = (float)s * exp2f(-(float)(2 * t + 1) * kscale);
  float s0, c0, s1, c1;
  sincosf(a0, &s0, &c0);   // accurate range reduction (angles up to ~4095 rad)
  sincosf(a1, &s1, &c1);
  const f32x4 pe = {s0, c0, s1, c1};

  // ---- batch loop, 3-stage software pipeline through LDS ----
#pragma unroll
  for (int b = 0; b < PB; ++b) {
    if (b + 2 < PB)
      async_load16(x + row + (size_t)(b + 2) * bstride, &buf[(b + 2) % 3][d0]);
    wait_async_n((PB - 1 - b) < 2 ? (PB - 1 - b) : 2);  // load b retired (in-order)
    f32x4 v = *(const f32x4*)&buf[b % 3][d0];  // ds_load_b128, thread-private region
    v += pe;
    __builtin_nontemporal_store(v, (f32x4*)(out + row + (size_t)b * bstride));
  }
}

extern "C" void kernel_launch(void* const* d_in, const int* in_sizes, int n_in,
                              void* d_out, int out_size, void* d_ws, size_t ws_size,
                              hipStream_t stream) {
  (void)in_sizes; (void)n_in; (void)d_ws; (void)ws_size; (void)out_size;
  const float* x = (const float*)d_in[0];
  float* out = (float*)d_out;
  pe_add_kernel<<<dim3(PS), dim3(256), 0, stream>>>(x, out);
}